// BRISQUE_43508018708576
// MI455X (gfx1250) — compile-verified
//
#include <hip/hip_runtime.h>
#include <hip/hip_bf16.h>
#include <math.h>

// ---------------------------------------------------------------------------
// BRISQUE pipeline for MI455X (gfx1250, wave32).
// Memory-bound problem (~700MB traffic => ~30us at 23.3TB/s). Resize is done
// as sparse 10-tap gathers (luma resides in 192MB L2). The only dense GEMM
// (16x36 features x 600 SVs) uses V_WMMA_F32_16X16X4_F32. MSCN tile loads use
// the CDNA5 async global->LDS path when available.
// ---------------------------------------------------------------------------

#define N_IMG 16
#define H0 1080
#define W0 1920
#define H1 540
#define W1 960
#define NPIX0 (H0 * W0)        // 2,073,600
#define NPIX1 (H1 * W1)        // 518,400
#define NSV 600
#define NSVP 608               // padded to 38 * 16
#define NCHUNK 38
#define NG 9801                // gamma table entries
#define NB0 135                // stat blocks per image, full res (1080/8)
#define NB1 68                 // stat blocks per image, half res

typedef float v2f __attribute__((ext_vector_type(2)));
typedef float v8f __attribute__((ext_vector_type(8)));

#if defined(__has_builtin)
#if __has_builtin(__builtin_amdgcn_global_load_async_to_lds_b32)
#define HAS_ASYNC_LDS 1
#endif
#endif

typedef __attribute__((address_space(1))) int* gptr_i32;
typedef __attribute__((address_space(3))) int* lptr_i32;

__device__ __forceinline__ void async_wait0() {
#if defined(__has_builtin)
#if __has_builtin(__builtin_amdgcn_s_wait_asynccnt)
  __builtin_amdgcn_s_wait_asynccnt(0);
#else
  asm volatile("s_wait_asynccnt 0x0" ::: "memory");
#endif
#else
  asm volatile("s_wait_asynccnt 0x0" ::: "memory");
#endif
}

#ifdef HAS_ASYNC_LDS
__device__ __forceinline__ void async_copy_b32(const float* g, float* l) {
  __builtin_amdgcn_global_load_async_to_lds_b32(
      (gptr_i32)(int*)(g), (lptr_i32)(int*)(l), 0, 0);
}
#endif

// FEATURE_RANGES lo/hi (36 features)
__constant__ float c_lo[36] = {
    0.338f, 0.017204f, 0.236f, -0.123884f, 0.000155f, 0.001122f,
    0.244f, -0.123586f, 0.000152f, 0.000975f, 0.249f, -0.135687f,
    0.000174f, 0.000913f, 0.258f, -0.143408f, 0.000179f, 0.000888f,
    0.471f, 0.012809f, 0.218f, -0.094876f, 1.5e-05f, 0.001272f,
    0.222f, -0.115772f, 1.6e-05f, 0.001374f, 0.227f, -0.117188f,
    3e-05f, 0.001122f, 0.228f, -0.12243f, 2.8e-05f, 0.001118f};
__constant__ float c_hi[36] = {
    10.f, 0.806612f, 1.642f, 0.20293f, 0.712298f, 0.470257f,
    1.641f, 0.179083f, 0.710456f, 0.470984f, 1.555f, 0.100858f,
    0.684173f, 0.534174f, 1.561f, 0.100486f, 0.685696f, 0.536508f,
    3.264f, 0.703171f, 1.046f, 0.187459f, 0.442057f, 0.40803f,
    1.042f, 0.162604f, 0.444362f, 0.40243f, 0.996f, 0.098323f,
    0.531903f, 0.369589f, 0.99f, 0.098658f, 0.530092f, 0.370399f};

// ---------------------------------------------------------------------------
// Tables: gamma-ratio table, bicubic half-resize gather tables, padded SVs
// ---------------------------------------------------------------------------
__device__ __forceinline__ double cubic_d(double x) {
  double a = fabs(x), a2 = a * a, a3 = a2 * a;
  if (a <= 1.0) return 1.5 * a3 - 2.5 * a2 + 1.0;
  if (a <= 2.0) return -0.5 * a3 + 2.5 * a2 - 4.0 * a + 2.0;
  return 0.0;
}

__device__ void make_resize_row(int o, int in_len, int* iX, float* wX) {
  // scale = 0.5, kernel width 8, P = 10 taps, mirror boundary
  double u = 2.0 * (double)(o + 1) - 0.5;
  double left = floor(u - 4.0);
  double wv[10], wsum = 0.0;
  for (int p = 0; p < 10; ++p) {
    wv[p] = 0.5 * cubic_d((u - (left + (double)p)) * 0.5);
    wsum += wv[p];
  }
  long twoN = 2 * (long)in_len;
  for (int p = 0; p < 10; ++p) {
    long ind = (long)left + p;
    long m = ((ind - 1) % twoN + twoN) % twoN;
    int id = (m < in_len) ? (int)m : (int)(twoN - 1 - m);
    iX[o * 10 + p] = id;
    wX[o * 10 + p] = (float)(wv[p] / wsum);
  }
}

__global__ void k_tables(float* __restrict__ rtab, int* __restrict__ iH,
                         float* __restrict__ wH, int* __restrict__ iW,
                         float* __restrict__ wW, const float* __restrict__ sv,
                         const float* __restrict__ coef,
                         float* __restrict__ svp, float* __restrict__ coefp,
                         float* __restrict__ normSV) {
  int t = blockIdx.x * blockDim.x + threadIdx.x;
  if (t < NG) {
    float g = (float)(0.2 + 0.001 * (double)t);
    rtab[t] = expf(2.f * lgammaf(2.f / g) - lgammaf(1.f / g) - lgammaf(3.f / g));
  }
  if (t < H1) make_resize_row(t, H0, iH, wH);
  if (t < W1) make_resize_row(t, W0, iW, wW);
  if (t < NSVP) {
    float s = 0.f;
    for (int k = 0; k < 36; ++k) {
      float v = (t < NSV) ? sv[t * 36 + k] : 0.f;
      svp[(size_t)t * 36 + k] = v;
      s += v * v;
    }
    normSV[t] = s;
    coefp[t] = (t < NSV) ? coef[t] : 0.f;
  }
}

// ---------------------------------------------------------------------------
// Luma: y = (0.299 r + 0.587 g + 0.114 b) * 255, float4 vectorized
// ---------------------------------------------------------------------------
__global__ void k_luma(const float* __restrict__ x, float* __restrict__ y) {
  int n = blockIdx.y;
  int i = (blockIdx.x * 256 + threadIdx.x) * 4;
  if (i >= NPIX0) return;
  const float* base = x + (size_t)n * 3 * NPIX0;
  float4 r = *(const float4*)(base + i);
  float4 g = *(const float4*)(base + NPIX0 + i);
  float4 b = *(const float4*)(base + 2 * NPIX0 + i);
  float4 o;
  o.x = (0.299f * r.x + 0.587f * g.x + 0.114f * b.x) * 255.f;
  o.y = (0.299f * r.y + 0.587f * g.y + 0.114f * b.y) * 255.f;
  o.z = (0.299f * r.z + 0.587f * g.z + 0.114f * b.z) * 255.f;
  o.w = (0.299f * r.w + 0.587f * g.w + 0.114f * b.w) * 255.f;
  *(float4*)(y + (size_t)n * NPIX0 + i) = o;
}

// ---------------------------------------------------------------------------
// MSCN: separable 7x7 Gaussian over y and y^2 in LDS tiles, zero-padded SAME.
// Interior tiles stream the halo tile straight to LDS via async copies.
// ---------------------------------------------------------------------------
__global__ void k_mscn(const float* __restrict__ y, float* __restrict__ xn,
                       int H, int W) {
  __shared__ float sy[38][40];
  __shared__ float sy2[38][40];
  __shared__ float hy[38][33];
  __shared__ float hy2[38][33];
  int n = blockIdx.z;
  int tx0 = blockIdx.x * 32, ty0 = blockIdx.y * 32;
  const float* img = y + (size_t)n * H * W;
  int tid = threadIdx.x;

  float g[7];
  {
    double tv[7], s = 0.0;
    const double twoSig2 = 2.0 * (7.0 / 6.0) * (7.0 / 6.0);
    for (int i = 0; i < 7; ++i) {
      double ax = (double)i - 3.0;
      tv[i] = exp(-ax * ax / twoSig2);
      s += tv[i];
    }
    for (int i = 0; i < 7; ++i) g[i] = (float)(tv[i] / s);
  }

#ifdef HAS_ASYNC_LDS
  bool interior = (tx0 >= 3) && (ty0 >= 3) && (tx0 + 35 <= W) && (ty0 + 35 <= H);
  if (interior) {
    const float* base = img + (size_t)(ty0 - 3) * W + (tx0 - 3);
    for (int i = tid; i < 38 * 38; i += 256) {
      int r = i / 38, c = i % 38;
      async_copy_b32(base + (size_t)r * W + c, &sy[r][c]);
    }
    async_wait0();  // this wave's async copies have landed in LDS
    for (int i = tid; i < 38 * 38; i += 256) {
      int r = i / 38, c = i % 38;
      float v = sy[r][c];
      sy2[r][c] = v * v;
    }
  } else
#endif
  {
    for (int i = tid; i < 38 * 38; i += 256) {
      int r = i / 38, c = i % 38;
      int h = ty0 - 3 + r, w = tx0 - 3 + c;
      float v = (h >= 0 && h < H && w >= 0 && w < W) ? img[(size_t)h * W + w] : 0.f;
      sy[r][c] = v;
      sy2[r][c] = v * v;
    }
  }
  __syncthreads();
  for (int i = tid; i < 38 * 32; i += 256) {
    int r = i / 32, c = i % 32;
    float a = 0.f, b = 0.f;
#pragma unroll
    for (int j = 0; j < 7; ++j) {
      a += g[j] * sy[r][c + j];
      b += g[j] * sy2[r][c + j];
    }
    hy[r][c] = a;
    hy2[r][c] = b;
  }
  __syncthreads();
  for (int i = tid; i < 32 * 32; i += 256) {
    int r = i / 32, c = i % 32;
    int h = ty0 + r, w = tx0 + c;
    if (h < H && w < W) {
      float mu = 0.f, m2 = 0.f;
#pragma unroll
      for (int j = 0; j < 7; ++j) {
        mu += g[j] * hy[r + j][c];
        m2 += g[j] * hy2[r + j][c];
      }
      float sig = sqrtf(fabsf(m2 - mu * mu) + 1e-8f);
      xn[(size_t)n * H * W + (size_t)h * W + w] = (sy[r + 3][c + 3] - mu) / (sig + 1.f);
    }
  }
}

// ---------------------------------------------------------------------------
// Statistics: 26 reductions per image (GGD: 2; 4 shifts x 6 AGGD sums)
// Deterministic: block partials -> fixed-order reduce kernel
// ---------------------------------------------------------------------------
__device__ __forceinline__ void aggd_acc(float* L, float q) {
  if (q < 0.f) {
    L[0] += 1.f;
    L[2] += q * q;
  } else if (q > 0.f) {
    L[1] += 1.f;
    L[3] += q * q;
  }
  L[4] += fabsf(q);
  L[5] += q * q;
}

__global__ void k_stats(const float* __restrict__ xn, float* __restrict__ pstat,
                        int H, int W) {
  int n = blockIdx.y;
  int NB = gridDim.x;
  const float* p = xn + (size_t)n * H * W;
  float loc[26];
#pragma unroll
  for (int j = 0; j < 26; ++j) loc[j] = 0.f;

  int rpb = (H + NB - 1) / NB;
  int r0 = blockIdx.x * rpb;
  int r1 = min(H, r0 + rpb);
  for (int h = r0; h < r1; ++h) {
    int hm = (h == 0) ? H - 1 : h - 1;
    int hp = (h == H - 1) ? 0 : h + 1;
    const float* rc = p + (size_t)h * W;
    const float* rm = p + (size_t)hm * W;
    const float* rp = p + (size_t)hp * W;
    for (int w = threadIdx.x; w < W; w += 256) {
      int wm = (w == 0) ? W - 1 : w - 1;
      float v = rc[w];
      loc[0] += v * v;
      loc[1] += fabsf(v);
      aggd_acc(loc + 2, v * rc[wm]);   // shift (0,1)
      aggd_acc(loc + 8, v * rm[w]);    // shift (1,0)
      aggd_acc(loc + 14, v * rm[wm]);  // shift (1,1)
      aggd_acc(loc + 20, v * rp[wm]);  // shift (-1,1)
    }
  }

  __shared__ float red[256];
  for (int j = 0; j < 26; ++j) {
    red[threadIdx.x] = loc[j];
    __syncthreads();
    for (int s = 128; s > 0; s >>= 1) {
      if (threadIdx.x < s) red[threadIdx.x] += red[threadIdx.x + s];
      __syncthreads();
    }
    if (threadIdx.x == 0) pstat[((size_t)n * 26 + j) * NB + blockIdx.x] = red[0];
    __syncthreads();
  }
}

__global__ void k_stats_red(const float* __restrict__ pstat,
                            float* __restrict__ acc, int NB, int scale) {
  int t = blockIdx.x * blockDim.x + threadIdx.x;  // n*26 + j
  if (t >= N_IMG * 26) return;
  int n = t / 26, j = t % 26;
  float s = 0.f;
  for (int b = 0; b < NB; ++b) s += pstat[(size_t)t * NB + b];
  acc[scale * 512 + n * 32 + j] = s;
}

// ---------------------------------------------------------------------------
// Bicubic half resize: sparse 10-tap gathers (rows, then cols)
// ---------------------------------------------------------------------------
__global__ void k_rrows(const float* __restrict__ y, const int* __restrict__ iH,
                        const float* __restrict__ wH, float* __restrict__ t) {
  int n = blockIdx.z, o = blockIdx.y;
  int w = blockIdx.x * 256 + threadIdx.x;
  if (w >= W0) return;
  const float* img = y + (size_t)n * NPIX0;
  float s = 0.f;
#pragma unroll
  for (int p = 0; p < 10; ++p)
    s += wH[o * 10 + p] * img[(size_t)iH[o * 10 + p] * W0 + w];
  t[((size_t)n * H1 + o) * W0 + w] = s;
}

__global__ void k_rcols(const float* __restrict__ t, const int* __restrict__ iW,
                        const float* __restrict__ wW, float* __restrict__ yh) {
  int n = blockIdx.z, o = blockIdx.y;
  int q = blockIdx.x * 256 + threadIdx.x;
  if (q >= W1) return;
  const float* row = t + ((size_t)n * H1 + o) * W0;
  float s = 0.f;
#pragma unroll
  for (int p = 0; p < 10; ++p) s += wW[q * 10 + p] * row[iW[q * 10 + p]];
  yh[((size_t)n * H1 + o) * W1 + q] = s;
}

// ---------------------------------------------------------------------------
// Features: GGD/AGGD parameter lookup (argmin over 9801-entry table),
// feature scaling to [-1,1], per-image squared norms
// ---------------------------------------------------------------------------
__global__ void k_features(const float* __restrict__ acc,
                           const float* __restrict__ rtab,
                           float* __restrict__ sf, float* __restrict__ normF) {
  int n = blockIdx.x, tid = threadIdx.x;
  __shared__ float f[36];
  __shared__ float bv[256];
  __shared__ int bi[256];

  for (int scale = 0; scale < 2; ++scale) {
    const float* A = acc + scale * 512 + n * 32;
    float HW = scale ? (float)NPIX1 : (float)NPIX0;

    float targ[5], slv[4], srv[4];
    {
      float s2 = A[0] / HW, E = A[1] / HW;
      targ[0] = s2 / (E * E);
      for (int t = 0; t < 4; ++t) {
        const float* B = A + 2 + 6 * t;
        float sl = sqrtf(B[2] / B[0]);
        float sr = sqrtf(B[3] / B[1]);
        slv[t] = sl;
        srv[t] = sr;
        float gh = sl / sr;
        float mabs = B[4] / HW, m2 = B[5] / HW;
        float rhat = (mabs * mabs) / m2;
        float gh2 = gh * gh;
        targ[1 + t] = rhat * (gh2 * gh + 1.f) * (gh + 1.f) / ((gh2 + 1.f) * (gh2 + 1.f));
      }
    }

    int idxs[5];
    for (int t = 0; t < 5; ++t) {
      float best = 3.4e38f;
      int bid = 0;
      for (int i = tid; i < NG; i += 256) {
        float d = fabsf(rtab[i] - targ[t]);
        if (d < best) { best = d; bid = i; }
      }
      bv[tid] = best;
      bi[tid] = bid;
      __syncthreads();
      for (int s = 128; s > 0; s >>= 1) {
        if (tid < s) {
          if (bv[tid + s] < bv[tid] ||
              (bv[tid + s] == bv[tid] && bi[tid + s] < bi[tid])) {
            bv[tid] = bv[tid + s];
            bi[tid] = bi[tid + s];
          }
        }
        __syncthreads();
      }
      idxs[t] = bi[0];
      __syncthreads();
    }

    if (tid == 0) {
      f[scale * 18 + 0] = (float)(0.2 + 0.001 * (double)idxs[0]);
      f[scale * 18 + 1] = A[0] / HW;  // s^2
      for (int t = 0; t < 4; ++t) {
        float a = (float)(0.2 + 0.001 * (double)idxs[1 + t]);
        float eta = (srv[t] - slv[t]) *
                    expf(lgammaf(2.f / a) - 0.5f * (lgammaf(1.f / a) + lgammaf(3.f / a)));
        int b = scale * 18 + 2 + 4 * t;
        f[b] = a;
        f[b + 1] = eta;
        f[b + 2] = slv[t] * slv[t];
        f[b + 3] = srv[t] * srv[t];
      }
    }
    __syncthreads();
  }

  __shared__ float nrm[36];
  if (tid < 36) {
    float v = -1.f + 2.f * (f[tid] - c_lo[tid]) / (c_hi[tid] - c_lo[tid]);
    sf[n * 40 + tid] = v;
    nrm[tid] = v * v;
  }
  if (tid >= 36 && tid < 40) sf[n * 40 + tid] = 0.f;
  __syncthreads();
  if (tid == 0) {
    float s = 0.f;
    for (int j = 0; j < 36; ++j) s += nrm[j];
    normF[n] = s;
  }
}

// ---------------------------------------------------------------------------
// RBF-SVM via WMMA f32 16x16x4: dot(features, sv^T) per 16-SV tile.
// A (16x4): VGPR0 holds K={k0,k0+2} split by lane half, VGPR1 K={k0+1,k0+3}.
// B (4x16): same K pairing, N striped across 16 lanes. One wave per tile.
// SVs come from the zero-padded svp buffer => unconditional loads, no EXEC
// churn between the 9 chained WMMAs.
// ---------------------------------------------------------------------------
__global__ void k_svm(const float* __restrict__ sf, const float* __restrict__ normF,
                      const float* __restrict__ svp, const float* __restrict__ coefp,
                      const float* __restrict__ normSV, float* __restrict__ pbuf) {
  int chunk = blockIdx.x;          // 0..37
  int lane = threadIdx.x;          // 0..31
  int half = lane >> 4;
  int l16 = lane & 15;
  int s = chunk * 16 + l16;        // SV index within padded 608

  const float* arow = sf + l16 * 40;
  const float* brow = svp + (size_t)s * 36;

  v8f c = {};
#pragma unroll
  for (int kk = 0; kk < 9; ++kk) {
    int k0 = kk * 4 + 2 * half;
    v2f a, b;
    a.x = arow[k0];
    a.y = arow[k0 + 1];
    b.x = brow[k0];
    b.y = brow[k0 + 1];
    c = __builtin_amdgcn_wmma_f32_16x16x4_f32(false, a, false, b, (short)0, c,
                                              false, false);
  }

  float cf = coefp[s];
  float ns = normSV[s];
#pragma unroll
  for (int r = 0; r < 8; ++r) {
    int m = r + 8 * half;          // image index held by this VGPR/lane-half
    float dist = normF[m] + ns - 2.f * c[r];
    float val = expf(-0.05f * dist) * cf;
    // deterministic reduction over the 16 lanes of this half (distinct SVs)
    for (int off = 8; off > 0; off >>= 1) val += __shfl_xor(val, off, 16);
    if (l16 == 0) pbuf[chunk * 16 + m] = val;
  }
}

__global__ void k_finish(const float* __restrict__ pbuf, float* __restrict__ out) {
  int m = threadIdx.x;
  if (m < N_IMG) {
    float s = 153.591f;
    for (int cnk = 0; cnk < NCHUNK; ++cnk) s += pbuf[cnk * 16 + m];
    out[m] = s;
  }
}

// ---------------------------------------------------------------------------
// Host launch
// ---------------------------------------------------------------------------
extern "C" void kernel_launch(void* const* d_in, const int* in_sizes, int n_in,
                              void* d_out, int out_size, void* d_ws, size_t ws_size,
                              hipStream_t stream) {
  const float* x = (const float*)d_in[0];    // [16,3,1080,1920]
  const float* sv = (const float*)d_in[1];   // [600,36]
  const float* coef = (const float*)d_in[2]; // [600]
  float* out = (float*)d_out;                // [16]
  float* ws = (float*)d_ws;

  // workspace layout (float element offsets); total ~266 MB
  float* rtab = ws;                       // 10240
  float* acc = ws + 10240;                // 1024 (26 stats x 16 imgs x 2 scales)
  float* sfb = ws + 11264;                // 16 x 40
  float* normF = ws + 11904;              // 16
  float* normSV = ws + 11920;             // 640
  float* wH = ws + 12560;                 // 5408
  float* wW = ws + 17968;                 // 9600
  int* iH = (int*)(ws + 27568);           // 5408
  int* iW = (int*)(ws + 32976);           // 9600
  float* pbuf = ws + 42576;               // 608 (SVM tile partials)
  float* pstat = ws + 43184;              // 16*26*135 = 56160 -> ends 99344
  float* svp = ws + 99344;                // 608*36 = 21888 -> ends 121232
  float* coefp = ws + 121232;             // 608
  float* y = ws + 131072;                 // 33,177,600 (full-res luma)
  float* B = y + (size_t)N_IMG * NPIX0;   // 33,177,600 shared region:
  float* xn = B;                          //   full-res MSCN (consumed by stats)
  float* tmp = B;                         //   then: row-resized [16,540,1920]
  float* yh = B + (size_t)N_IMG * H1 * W0;        // half-res luma [16,540,960]
  float* xnh = yh + (size_t)N_IMG * NPIX1;        // half-res MSCN

  k_tables<<<dim3((NG + 255) / 256), 256, 0, stream>>>(rtab, iH, wH, iW, wW, sv,
                                                       coef, svp, coefp, normSV);
  k_luma<<<dim3(NPIX0 / 4 / 256, N_IMG), 256, 0, stream>>>(x, y);

  // full resolution: MSCN -> stats
  k_mscn<<<dim3(W0 / 32, (H0 + 31) / 32, N_IMG), 256, 0, stream>>>(y, xn, H0, W0);
  k_stats<<<dim3(NB0, N_IMG), 256, 0, stream>>>(xn, pstat, H0, W0);
  k_stats_red<<<dim3(2), 256, 0, stream>>>(pstat, acc, NB0, 0);

  // half resolution: resize (reuses region B after stats consumed xn)
  k_rrows<<<dim3((W0 + 255) / 256, H1, N_IMG), 256, 0, stream>>>(y, iH, wH, tmp);
  k_rcols<<<dim3((W1 + 255) / 256, H1, N_IMG), 256, 0, stream>>>(tmp, iW, wW, yh);
  k_mscn<<<dim3(W1 / 32, (H1 + 31) / 32, N_IMG), 256, 0, stream>>>(yh, xnh, H1, W1);
  k_stats<<<dim3(NB1, N_IMG), 256, 0, stream>>>(xnh, pstat, H1, W1);
  k_stats_red<<<dim3(2), 256, 0, stream>>>(pstat, acc, NB1, 1);

  // features + SVM (WMMA f32 16x16x4) + deterministic finish
  k_features<<<dim3(N_IMG), 256, 0, stream>>>(acc, rtab, sfb, normF);
  k_svm<<<dim3(NCHUNK), 32, 0, stream>>>(sfb, normF, svp, coefp, normSV, pbuf);
  k_finish<<<dim3(1), 32, 0, stream>>>(pbuf, out);
}